// DuplicateByDuration_50697793962727
// MI455X (gfx1250) — compile-verified
//
#include <hip/hip_runtime.h>
#include <stdint.h>

// Problem dims (fixed by the reference)
#define B_      16
#define C_      192
#define TT      512     // T_TEXT
#define TF      2048    // T_FEAT
#define CH      16      // channels per block in gather kernel
#define K2T     256     // threads in gather kernel

// ---- CDNA5 async global->LDS copy (gfx1250), tracked by ASYNCcnt ----
__device__ __forceinline__ void async_copy_b128(uint32_t lds_byte_off, const void* gptr) {
    // VDST = VGPR holding workgroup-relative LDS byte address,
    // VADDR = 64-bit global address, GV mode ("off")
    asm volatile("global_load_async_to_lds_b128 %0, %1, off"
                 :: "v"(lds_byte_off), "v"(gptr)
                 : "memory");
}
__device__ __forceinline__ void wait_async_all() {
    asm volatile("s_wait_asynccnt 0" ::: "memory");
}
__device__ __forceinline__ uint32_t lds_off_of(const void* p) {
    // low 32 bits of a generic pointer into LDS == workgroup-relative LDS offset
    return (uint32_t)(uintptr_t)p;
}

// =====================================================================
// Kernel 1: per batch, build per-frame token index map idx[b][j]
//   idx = i such that cum[i-1] <= j < cum[i], else -1
// =====================================================================
__global__ __launch_bounds__(TT)
void build_idx_kernel(const int* __restrict__ w,
                      const float* __restrict__ xmask,
                      int* __restrict__ gidx) {
    const int b   = blockIdx.x;
    const int tid = threadIdx.x;

    __shared__ int scum[TT];
    __shared__ int sidx[TF];

    // masked duration, truncate-toward-zero like (w.f32 * mask).astype(i32)
    int wi = (int)((float)w[b * TT + tid] * xmask[b * TT + tid]);
    if (wi < 0) wi = 0;
    scum[tid] = wi;
    __syncthreads();

    // inclusive scan (Hillis-Steele) over 512 elements
    for (int off = 1; off < TT; off <<= 1) {
        int add = (tid >= off) ? scum[tid - off] : 0;
        __syncthreads();
        scum[tid] += add;
        __syncthreads();
    }

    // init frame map to -1 (frames past total duration -> zero output)
    for (int j = tid; j < TF; j += TT) sidx[j] = -1;
    __syncthreads();

    // scatter: token tid owns frames [cum-wi, cum)
    int e0 = scum[tid];
    int s0 = e0 - wi;
    if (s0 < 0)  s0 = 0;
    if (s0 > TF) s0 = TF;
    if (e0 > TF) e0 = TF;
    for (int j = s0; j < e0; ++j) sidx[j] = tid;
    __syncthreads();

    // coalesced write-out
    for (int j = tid; j < TF; j += TT) gidx[b * TF + j] = sidx[j];
}

// =====================================================================
// Kernel 2: gather/duplicate. One block = (batch b, 16-channel tile).
// Stage x-tile, x_mask, idx, y_mask into LDS via async b128 copies,
// hoist frame-indexed values once per frame-quad, then emit float4
// coalesced stores (4 LDS gathers + 1 b128 store per channel-quad).
// =====================================================================
__global__ __launch_bounds__(K2T)
void dup_gather_kernel(const float* __restrict__ x,
                       const float* __restrict__ xmask,
                       const float* __restrict__ ymask,
                       const int*   __restrict__ gidx,
                       float* __restrict__ y) {
    const int ct  = blockIdx.x;      // channel tile: 0..(C_/CH-1)
    const int b   = blockIdx.y;      // batch
    const int tid = threadIdx.x;     // 0..255

    __shared__ __align__(16) float sx[CH * TT];   // 32 KB: x tile
    __shared__ __align__(16) float sm[TT];        //  2 KB: x_mask row
    __shared__ __align__(16) int   sidx[TF];      //  8 KB: frame->token map
    __shared__ __align__(16) float sym[TF];       //  8 KB: y_mask row

    const float* xb = x     + ((size_t)b * C_ + (size_t)ct * CH) * TT;
    const float* xm = xmask + (size_t)b * TT;
    const float* ym = ymask + (size_t)b * TF;
    const int*   gi = gidx  + (size_t)b * TF;

    const uint32_t sx_base = lds_off_of(sx);
    const uint32_t sm_base = lds_off_of(sm);
    const uint32_t si_base = lds_off_of(sidx);
    const uint32_t sy_base = lds_off_of(sym);

    // x tile: CH*TT = 8192 floats -> 2048 b128 transfers (8 per thread)
    #pragma unroll
    for (int e = tid * 4; e < CH * TT; e += K2T * 4)
        async_copy_b128(sx_base + (uint32_t)e * 4u, xb + e);
    // x_mask row: 512 floats
    for (int e = tid * 4; e < TT; e += K2T * 4)
        async_copy_b128(sm_base + (uint32_t)e * 4u, xm + e);
    // idx row: 2048 ints
    #pragma unroll
    for (int e = tid * 4; e < TF; e += K2T * 4)
        async_copy_b128(si_base + (uint32_t)e * 4u, gi + e);
    // y_mask row: 2048 floats
    #pragma unroll
    for (int e = tid * 4; e < TF; e += K2T * 4)
        async_copy_b128(sy_base + (uint32_t)e * 4u, ym + e);

    wait_async_all();     // this wave's async copies landed in LDS
    __syncthreads();      // all waves' copies visible

    float* yb = y + ((size_t)b * C_ + (size_t)ct * CH) * TF;

    #pragma unroll
    for (int jb = 0; jb < TF / (K2T * 4); ++jb) {   // 2 frame-quads per thread
        const int j0 = (jb * K2T + tid) * 4;

        // hoisted per-frame values (shared by all 16 channels)
        int   ii[4];
        float sc[4];
        #pragma unroll
        for (int k = 0; k < 4; ++k) {
            int t = sidx[j0 + k];
            float s = (t >= 0) ? sm[t] * sym[j0 + k] : 0.0f;
            ii[k] = (t >= 0) ? t : 0;   // safe gather index; sc==0 kills it
            sc[k] = s;
        }

        #pragma unroll
        for (int c = 0; c < CH; ++c) {
            const float* xrow = &sx[c * TT];
            float4 o;
            float* op = (float*)&o;
            #pragma unroll
            for (int k = 0; k < 4; ++k)
                op[k] = xrow[ii[k]] * sc[k];
            *reinterpret_cast<float4*>(yb + (size_t)c * TF + j0) = o;
        }
    }
}

// =====================================================================
extern "C" void kernel_launch(void* const* d_in, const int* in_sizes, int n_in,
                              void* d_out, int out_size, void* d_ws, size_t ws_size,
                              hipStream_t stream) {
    (void)in_sizes; (void)n_in; (void)out_size; (void)ws_size;

    const float* x     = (const float*)d_in[0];   // (B, C, T_text)
    const int*   w     = (const int*)  d_in[1];   // (B, T_text)
    const float* xmask = (const float*)d_in[2];   // (B, T_text)
    const float* ymask = (const float*)d_in[3];   // (B, T_feat)
    float*       out   = (float*)d_out;           // (B, C, T_feat)

    int* gidx = (int*)d_ws;                       // B*TF ints = 128 KB scratch

    build_idx_kernel<<<dim3(B_), dim3(TT), 0, stream>>>(w, xmask, gidx);

    dup_gather_kernel<<<dim3(C_ / CH, B_), dim3(K2T), 0, stream>>>(
        x, xmask, ymask, gidx, out);
}